// FAST_28071906247131
// MI455X (gfx1250) — compile-verified
//
#include <hip/hip_runtime.h>

#define TPB 256   // 8 wave32 per workgroup
#define TPT 4     // triangles per thread -> 48 B/thread, b128-aligned streams

typedef float v4f __attribute__((ext_vector_type(4)));  // native vector: valid for nontemporal builtins

__device__ __forceinline__ float edge_len(float ax, float ay, float az,
                                          float bx, float by, float bz) {
    float dx = ax - bx, dy = ay - by, dz = az - bz;
    return __builtin_amdgcn_sqrtf(fmaf(dx, dx, fmaf(dy, dy, dz * dz)));
}

__global__ __launch_bounds__(TPB) void fast_edge_kernel(
    const float* __restrict__ pos,   // [N,3] node coords (2.4 MB, L2-resident)
    const int*   __restrict__ tri,   // [T,3] vertex indices (streamed)
    float*       __restrict__ out,   // [T,3] edge lengths (streamed)
    int nTri)
{
    // This block's slice of the triangle index stream: 256 * 48 B = 12 KB LDS.
    __shared__ alignas(16) int stri[TPB * 3 * TPT];

    const long gid = (long)blockIdx.x * TPB + threadIdx.x;
    const long t0  = gid * TPT;
    const int  l   = threadIdx.x * (3 * TPT);

    if (t0 + TPT <= (long)nTri) {
        const unsigned goff  = (unsigned)(t0 * 3u * sizeof(int));  // byte offset into tri[]
        // Low 32 bits of the flat pointer == LDS byte address (ISA 10.2 aperture mapping).
        const unsigned laddr = (unsigned)(uintptr_t)&stri[l];

        // Speculative L2 prefetch of the triangle stream ~256 KB ahead (global_prefetch_b8).
        __builtin_prefetch((const char*)tri + goff + (256u << 10), 0, 1);

        // CDNA5 async global->LDS copy of this thread's 48 B of indices (3 x b128),
        // tracked by ASYNCcnt; no VGPR return traffic. offset: applies to both sides.
        asm volatile(
            "global_load_async_to_lds_b128 %0, %1, %2\n\t"
            "global_load_async_to_lds_b128 %0, %1, %2 offset:16\n\t"
            "global_load_async_to_lds_b128 %0, %1, %2 offset:32\n\t"
            "s_wait_asynccnt 0"
            :
            : "v"(laddr), "v"(goff), "s"(tri)
            : "memory");

        float r[3 * TPT];
#pragma unroll
        for (int k = 0; k < TPT; ++k) {
            const int i0 = stri[l + 3 * k + 0];
            const int i1 = stri[l + 3 * k + 1];
            const int i2 = stri[l + 3 * k + 2];
            const float x0 = pos[3 * i0 + 0], y0 = pos[3 * i0 + 1], z0 = pos[3 * i0 + 2];
            const float x1 = pos[3 * i1 + 0], y1 = pos[3 * i1 + 1], z1 = pos[3 * i1 + 2];
            const float x2 = pos[3 * i2 + 0], y2 = pos[3 * i2 + 1], z2 = pos[3 * i2 + 2];
            r[3 * k + 0] = edge_len(x0, y0, z0, x1, y1, z1);  // |v0-v1|
            r[3 * k + 1] = edge_len(x1, y1, z1, x2, y2, z2);  // |v1-v2|
            r[3 * k + 2] = edge_len(x2, y2, z2, x0, y0, z0);  // |v2-v0|
        }

        // Streaming non-temporal b128 stores: keep the positions table resident in L2.
        v4f* op = (v4f*)(out + t0 * 3);
        v4f v0 = {r[0], r[1], r[2],  r[3]};
        v4f v1 = {r[4], r[5], r[6],  r[7]};
        v4f v2 = {r[8], r[9], r[10], r[11]};
        __builtin_nontemporal_store(v0, op + 0);
        __builtin_nontemporal_store(v1, op + 1);
        __builtin_nontemporal_store(v2, op + 2);
    } else if (t0 < (long)nTri) {
        // Scalar tail (last partial block): direct global loads, per-triangle.
        for (long t = t0; t < (long)nTri; ++t) {
            const int i0 = tri[3 * t + 0];
            const int i1 = tri[3 * t + 1];
            const int i2 = tri[3 * t + 2];
            const float x0 = pos[3 * i0 + 0], y0 = pos[3 * i0 + 1], z0 = pos[3 * i0 + 2];
            const float x1 = pos[3 * i1 + 0], y1 = pos[3 * i1 + 1], z1 = pos[3 * i1 + 2];
            const float x2 = pos[3 * i2 + 0], y2 = pos[3 * i2 + 1], z2 = pos[3 * i2 + 2];
            out[3 * t + 0] = edge_len(x0, y0, z0, x1, y1, z1);
            out[3 * t + 1] = edge_len(x1, y1, z1, x2, y2, z2);
            out[3 * t + 2] = edge_len(x2, y2, z2, x0, y0, z0);
        }
    }
}

extern "C" void kernel_launch(void* const* d_in, const int* in_sizes, int n_in,
                              void* d_out, int out_size, void* d_ws, size_t ws_size,
                              hipStream_t stream) {
    const float* pos = (const float*)d_in[0];   // positions: N_NODES*3 floats
    const int*   tri = (const int*)d_in[1];     // triangles: N_TRIS*3 ints
    float*       out = (float*)d_out;           // paddings:  N_TRIS*3 floats

    const int nTri    = in_sizes[1] / 3;
    const int threads = (nTri + TPT - 1) / TPT;
    const int blocks  = (threads + TPB - 1) / TPB;

    fast_edge_kernel<<<blocks, TPB, 0, stream>>>(pos, tri, out, nTri);
}